// LocalDetailAttention_71073118814383
// MI455X (gfx1250) — compile-verified
//
#include <hip/hip_runtime.h>

typedef float v2f __attribute__((ext_vector_type(2)));
typedef float v8f __attribute__((ext_vector_type(8)));

#define B_ 2
#define L_ 4096
#define H_ 8
#define E_ 64

#define BAND_ROWS 128   // union key band for 64 queries: [q0-32, q0+95]
#define BSTR 76         // band row stride (floats): bank-conflict-free pad
#define ASTR 84         // dense-score row stride (floats): 20r+c tiles 64 banks

__global__ __launch_bounds__(128)
void lda_wmma_kernel(const float* __restrict__ Q,
                     const float* __restrict__ K,
                     const float* __restrict__ V,
                     float* __restrict__ O)
{
    __shared__ float band[BAND_ROWS * BSTR];   // 38,912 B: K band, then reused for V band
    __shared__ float ad[4 * 16 * ASTR];        // 21,504 B: per-wave dense scores/weights

    const int tid  = threadIdx.x;
    const int lane = tid & 31;
    const int wave = tid >> 5;
    const int lrow = lane & 15;   // M / N index within 16
    const int lhi  = lane >> 4;   // K-half selector (0: k=0,1  1: k=2,3)

    const int tile = blockIdx.x & 63;          // L/64 tiles
    const int h    = (blockIdx.x >> 6) & 7;
    const int b    = blockIdx.x >> 9;
    const int q0   = tile * 64;

    // ---------------- Q rows of this wave -> A-operand registers ----------------
    // A(16x4) layout: lanes 0-15 hold M=lane, k = {0,1}; lanes 16-31 hold M, k = {2,3}
    v2f qa[16];
    {
        const int qrow = q0 + wave * 16 + lrow;
        const float* qp = Q + (((size_t)b * L_ + qrow) * H_ + h) * E_ + 2 * lhi;
        #pragma unroll
        for (int kk = 0; kk < 16; ++kk) {
            qa[kk][0] = qp[4 * kk + 0];
            qa[kk][1] = qp[4 * kk + 1];
        }
    }

    // ---------------- cooperative load: K band (128 rows x 64 fp32) ----------------
    #pragma unroll
    for (int i = 0; i < 16; ++i) {
        const int f   = i * 128 + tid;
        const int row = f >> 4;               // 0..127
        const int c4  = (f & 15) << 2;        // 0..60, float4 column
        int p = q0 - 32 + row;                // clamp == reference's index clip
        p = p < 0 ? 0 : (p > (L_ - 1) ? (L_ - 1) : p);
        const float4 g = *(const float4*)(K + (((size_t)b * L_ + p) * H_ + h) * E_ + c4);
        *(float4*)(&band[row * BSTR + c4]) = g;
    }
    __syncthreads();

    // ---------------- GEMM1: S(16x80) = Q(16x64) * Kband^T ----------------
    // wave w uses band rows [16w, 16w+79]; column tile c covers band rows 16(w+c)..+15
    v8f sacc[5];
    #pragma unroll
    for (int c = 0; c < 5; ++c) sacc[c] = (v8f){0.f,0.f,0.f,0.f,0.f,0.f,0.f,0.f};
    #pragma unroll
    for (int kk = 0; kk < 16; ++kk) {
        #pragma unroll
        for (int c = 0; c < 5; ++c) {
            // B(4x16): lane holds col N=lrow, k-pair (2*lhi,2*lhi+1) -> contiguous float2 in band row
            const v2f bv = *(const v2f*)(&band[(16 * (wave + c) + lrow) * BSTR + 4 * kk + 2 * lhi]);
            sacc[c] = __builtin_amdgcn_wmma_f32_16x16x4_f32(
                false, qa[kk], false, bv, (short)0, sacc[c], false, false);
        }
    }

    // scatter dense scores: C/D layout -> ad[wave][M][j]
    float* aw = &ad[wave * 16 * ASTR];
    #pragma unroll
    for (int c = 0; c < 5; ++c) {
        #pragma unroll
        for (int r = 0; r < 8; ++r)
            aw[(r + 8 * lhi) * ASTR + 16 * c + lrow] = sacc[c][r];
    }
    __syncthreads();   // all waves finished reading K band; scores parked in LDS

    // ---------------- cooperative load: V band (overwrites K band) ----------------
    #pragma unroll
    for (int i = 0; i < 16; ++i) {
        const int f   = i * 128 + tid;
        const int row = f >> 4;
        const int c4  = (f & 15) << 2;
        int p = q0 - 32 + row;
        p = p < 0 ? 0 : (p > (L_ - 1) ? (L_ - 1) : p);
        const float4 g = *(const float4*)(V + (((size_t)b * L_ + p) * H_ + h) * E_ + c4);
        *(float4*)(&band[row * BSTR + c4]) = g;
    }

    // ---------------- softmax over the 33 valid diagonal entries per row ----------------
    if (lane < 16) {
        const int i = lane;                   // local query row; valid j = i + 2k, k=0..32
        float* rowp = aw + i * ASTR;
        const float scale = 0.125f;           // 1/sqrt(64)
        float m = -3.402823466e38f;
        #pragma unroll 1
        for (int k = 0; k < 33; ++k) m = fmaxf(m, rowp[i + 2 * k]);
        m *= scale;
        float sum = 0.f;
        #pragma unroll 1
        for (int k = 0; k < 33; ++k) {
            const float e = __expf(rowp[i + 2 * k] * scale - m);
            rowp[i + 2 * k] = e;
            sum += e;
        }
        const float inv = 1.f / sum;
        #pragma unroll 1
        for (int k = 0; k < 33; ++k) rowp[i + 2 * k] *= inv;
        // zero the dense positions outside the dilated window
        #pragma unroll 1
        for (int j = 0; j < 80; ++j) {
            const bool valid = (j >= i) && (j <= i + 64) && (((j - i) & 1) == 0);
            if (!valid) rowp[j] = 0.f;
        }
    }
    __syncthreads();   // V band + normalized weights ready

    // ---------------- GEMM2: O(16x64) = A(16x80) * Vband(80x64) ----------------
    v8f oacc[4];
    #pragma unroll
    for (int c = 0; c < 4; ++c) oacc[c] = (v8f){0.f,0.f,0.f,0.f,0.f,0.f,0.f,0.f};
    #pragma unroll
    for (int kk = 0; kk < 20; ++kk) {
        const v2f av = *(const v2f*)(&aw[lrow * ASTR + 4 * kk + 2 * lhi]);
        const int vr = 16 * wave + 4 * kk + 2 * lhi;
        #pragma unroll
        for (int c = 0; c < 4; ++c) {
            v2f bv;   // B(4x16): lane needs 2 floats from same column, adjacent band rows
            bv[0] = band[(vr    ) * BSTR + 16 * c + lrow];
            bv[1] = band[(vr + 1) * BSTR + 16 * c + lrow];
            oacc[c] = __builtin_amdgcn_wmma_f32_16x16x4_f32(
                false, av, false, bv, (short)0, oacc[c], false, false);
        }
    }

    // ---------------- store output ----------------
    const int orow0 = q0 + wave * 16 + 8 * lhi;
    #pragma unroll
    for (int r = 0; r < 8; ++r) {
        float* op = O + (((size_t)b * L_ + (orow0 + r)) * H_ + h) * E_ + lrow;
        #pragma unroll
        for (int c = 0; c < 4; ++c) op[16 * c] = oacc[c][r];
    }
}

extern "C" void kernel_launch(void* const* d_in, const int* in_sizes, int n_in,
                              void* d_out, int out_size, void* d_ws, size_t ws_size,
                              hipStream_t stream) {
    (void)in_sizes; (void)n_in; (void)out_size; (void)d_ws; (void)ws_size;
    const float* Q = (const float*)d_in[0];
    const float* K = (const float*)d_in[1];
    const float* V = (const float*)d_in[2];
    float* O = (float*)d_out;
    // grid = B * H * (L/64) = 2 * 8 * 64 = 1024 blocks, 128 threads (4 wave32)
    lda_wmma_kernel<<<dim3(1024), dim3(128), 0, stream>>>(Q, K, V, O);
}